// NormalMatchingCRPSLoss_75600014344216
// MI455X (gfx1250) — compile-verified
//
#include <hip/hip_runtime.h>
#include <hip/hip_bf16.h>
#include <math.h>

#define B_   8
#define D1_  4
#define D2_  32
#define D3_  32
#define S_   32
#define T_   32
#define D_TOT (D1_*D2_*D3_)   /* 4096 */
#define K_    (S_*T_)         /* 1024 */
#define SINK_ITERS 100
#define EPS_W 0.002f

typedef float v2f __attribute__((ext_vector_type(2)));
typedef float v8f __attribute__((ext_vector_type(8)));

__device__ __forceinline__ float raw_crps(float y) {
    const float INV_SQRT2        = 0.7071067811865476f;
    const float TWO_INV_SQRT_2PI = 0.7978845608028654f;
    const float INV_SQRT_PI      = 0.5641895835477563f;
    return y * erff(y * INV_SQRT2) + TWO_INV_SQRT_2PI * expf(-0.5f * y * y) + INV_SQRT_PI;
}

// ---------------- kernel 0: zero the cost accumulator ----------------
__global__ void k_zero(float* __restrict__ p, int n) {
    int i = blockIdx.x * blockDim.x + threadIdx.x;
    if (i < n) p[i] = 0.0f;
}

// ---------------- kernel 1: crps_raw + cost partial sums ----------------
// One block per (b,d1,d2); loops d3. 256 threads, 4 (s,t) elems each, t fastest
// (coalesced stores of R). Per-thread register accumulation over d3, then one
// atomicAdd per (thread,j) into cost[b,s,t] (1M atomics total).
__global__ __launch_bounds__(256) void k_crps_cost(
        const float* __restrict__ locs, const float* __restrict__ lss,
        const float* __restrict__ tgts, float* __restrict__ R,
        float* __restrict__ cost) {
    __shared__ float s_loc[S_], s_sig[S_], s_isg[S_], s_tgt[T_];
    int blk = blockIdx.x;                 // 0..1023
    int b  = blk >> 7;                    // / (D1_*D2_)
    int r  = blk & 127;
    int d1 = r >> 5, d2 = r & 31;
    int tid = threadIdx.x;
    size_t base3 = (((size_t)(b * D1_ + d1) * D2_ + d2) * D3_);
    float acc[4] = {0.f, 0.f, 0.f, 0.f};
    for (int d3 = 0; d3 < D3_; ++d3) {
        if (tid < 32) {
            size_t off = (base3 + d3) * 32 + tid;
            float l = locs[off], ls = lss[off];
            s_loc[tid] = l;
            s_sig[tid] = expf(ls);
            s_isg[tid] = expf(-ls);
            s_tgt[tid] = tgts[off];
        }
        __syncthreads();
        size_t d = (size_t)d1 * 1024 + (size_t)d2 * 32 + d3;
        float* Rrow = R + ((size_t)b * D_TOT + d) * K_;
        #pragma unroll
        for (int j = 0; j < 4; ++j) {
            int k = tid + 256 * j;
            int s = k >> 5, t = k & 31;
            float y  = (s_tgt[t] - s_loc[s]) * s_isg[s];
            float cr = raw_crps(y) * s_sig[s];
            Rrow[k] = cr;
            acc[j] += cr;
        }
        __syncthreads();
    }
    float* cb = cost + (size_t)b * K_;
    #pragma unroll
    for (int j = 0; j < 4; ++j) atomicAdd(&cb[tid + 256 * j], acc[j]);
}

// ---------------- kernel 2: log-domain Sinkhorn, one wave32 per batch ----------------
// lane = t. logK[s] register-resident (32 regs/lane). Row LSE = butterfly
// shuffles across lanes; column LSE = in-lane reduction over the 32 registers.
__global__ __launch_bounds__(32) void k_sinkhorn(const float* __restrict__ cost,
                                                 float* __restrict__ wgt) {
    int b = blockIdx.x, lane = threadIdx.x;
    const float* cb = cost + (size_t)b * K_;
    float c[S_], logK[S_], log_u[S_];
    float nrm2 = 0.f;
    #pragma unroll
    for (int s = 0; s < S_; ++s) {
        float v = cb[s * T_ + lane] * (1.0f / D_TOT);  // mean over 4096 d-slices
        c[s] = v;  nrm2 += v * v;
    }
    #pragma unroll
    for (int off = 16; off; off >>= 1) nrm2 += __shfl_xor(nrm2, off, 32);
    float scl = -1.0f / (sqrtf(nrm2) * EPS_W);         // logK = -(cost/norm)/eps
    #pragma unroll
    for (int s = 0; s < S_; ++s) logK[s] = c[s] * scl;

    const float log_ab = logf(1.0f / 32.0f);
    float log_v = 0.0f;
    for (int it = 0; it <= SINK_ITERS; ++it) {
        // log_u[s] = log_a - logsumexp_t(logK[s,:] + log_v[:])
        #pragma unroll
        for (int s = 0; s < S_; ++s) {
            float x = logK[s] + log_v;
            float m = x;
            #pragma unroll
            for (int off = 16; off; off >>= 1) m = fmaxf(m, __shfl_xor(m, off, 32));
            float e = expf(x - m);
            #pragma unroll
            for (int off = 16; off; off >>= 1) e += __shfl_xor(e, off, 32);
            log_u[s] = log_ab - (m + logf(e));
        }
        if (it == SINK_ITERS) break;  // final extra log_u update per reference
        // log_v[t] = log_b - logsumexp_s(logK[:,t] + log_u[:])
        float m = -1e30f;
        #pragma unroll
        for (int s = 0; s < S_; ++s) m = fmaxf(m, logK[s] + log_u[s]);
        float e = 0.f;
        #pragma unroll
        for (int s = 0; s < S_; ++s) e += expf(logK[s] + log_u[s] - m);
        log_v = log_ab - (m + logf(e));
    }
    float* wb = wgt + (size_t)b * K_;
    #pragma unroll
    for (int s = 0; s < S_; ++s) wb[s * T_ + lane] = expf(logK[s] + log_u[s] + log_v);
}

// ---------------- kernel 3: WMMA f32 GEMV contraction ----------------
// out[b,d] = sum_k R[b,d,k] * w[b,k]; one wave per 16-row tile; K-loop in
// steps of 4 via V_WMMA_F32_16X16X4_F32. B matrix = weight chunk broadcast
// across the N dim, so every D column holds the final dot product.
__global__ __launch_bounds__(32) void k_wsum(const float* __restrict__ R,
                                             const float* __restrict__ wgt,
                                             float* __restrict__ out) {
    __shared__ float s_w[K_];
    int tile = blockIdx.x;              // 0..2047
    int b  = tile >> 8;                 // / 256 tiles per batch
    int d0 = (tile & 255) << 4;         // tile * 16
    int lane = threadIdx.x;
    const float* wb = wgt + (size_t)b * K_;
    for (int i = lane; i < K_; i += 32) s_w[i] = wb[i];
    __syncthreads();

    int hv = lane >> 4;                 // half: K pair selector (0 -> K0/1, 1 -> K2/3)
    int mr = lane & 15;                 // row within tile
    const float* Arow = R + ((size_t)b * D_TOT + d0 + mr) * K_;
    v8f cacc = {0.f, 0.f, 0.f, 0.f, 0.f, 0.f, 0.f, 0.f};
    for (int k = 0; k < K_; k += 4) {
        v2f a, bm;
        a[0]  = Arow[k + 2 * hv];       // A 16x4: VGPR0 = K0/K2, VGPR1 = K1/K3
        a[1]  = Arow[k + 2 * hv + 1];
        bm[0] = s_w[k + 2 * hv];        // B 4x16: rows striped across lanes; value
        bm[1] = s_w[k + 2 * hv + 1];    //   uniform over N (GEMV broadcast)
        cacc = __builtin_amdgcn_wmma_f32_16x16x4_f32(
                   false, a, false, bm, (short)0, cacc, false, false);
    }
    // C: VGPR j -> row j (lanes 0-15) / row j+8 (lanes 16-31); take column n=0
    if (mr == 0) {
        float* ob = out + (size_t)b * D_TOT + d0 + 8 * hv;
        #pragma unroll
        for (int j = 0; j < 8; ++j) ob[j] = cacc[j];
    }
}

extern "C" void kernel_launch(void* const* d_in, const int* in_sizes, int n_in,
                              void* d_out, int out_size, void* d_ws, size_t ws_size,
                              hipStream_t stream) {
    (void)in_sizes; (void)n_in; (void)out_size; (void)ws_size;
    const float* locs = (const float*)d_in[0];
    const float* lss  = (const float*)d_in[1];
    const float* tgts = (const float*)d_in[2];
    float* out = (float*)d_out;

    char* ws = (char*)d_ws;
    size_t Rbytes = (size_t)B_ * D_TOT * K_ * sizeof(float);   // 134 MB < 192 MB L2
    float* R    = (float*)ws;
    float* cost = (float*)(ws + Rbytes);
    float* wgt  = cost + (size_t)B_ * K_;

    k_zero<<<(B_ * K_ + 255) / 256, 256, 0, stream>>>(cost, B_ * K_);
    k_crps_cost<<<B_ * D1_ * D2_, 256, 0, stream>>>(locs, lss, tgts, R, cost);
    k_sinkhorn<<<B_, 32, 0, stream>>>(cost, wgt);
    k_wsum<<<B_ * (D_TOT / 16), 32, 0, stream>>>(R, wgt, out);
}